// NTM_39891656245685
// MI455X (gfx1250) — compile-verified
//
#include <hip/hip_runtime.h>
#include <math.h>

// ---------------- types ----------------
typedef float v2f __attribute__((ext_vector_type(2)));
typedef float v8f __attribute__((ext_vector_type(8)));

#define NROWS 65536
#define MCOLS 256
#define HHEADS 4
#define EPSF 1e-8f

// ---------------- workspace layout (floats) ----------------
#define WS_MEM    ((size_t)0)                    // 65536*256
#define WS_INP    (WS_MEM + (size_t)NROWS*MCOLS) // 3072
#define WS_GATES  (WS_INP + 3072)                // 4096
#define WS_CTRL   (WS_GATES + 4096)              // 1024
#define WS_OW     (WS_CTRL + 1024)               // 774 -> 784
#define WS_ORR    (WS_OW + 784)                  // 262 -> 272
#define WS_PW     (WS_ORR + 272)                 // k(256)+e(256)+a(256)+sc(7) -> 784
#define WS_PR     (WS_PW + 784)                  // k(256)+sc(7) -> 272
#define WS_SIM    (WS_PR + 272)                  // 65536
#define WS_P      (WS_SIM + 65536)               // 65536
#define WS_WSH    (WS_P + 65536)                 // 65536
#define WS_WVEC   (WS_WSH + 65536)               // 65536
#define WS_BLK    (WS_WVEC + 65536)              // 1024
#define WS_SCAL   (WS_BLK + 1024)                // 16
#define WS_READS  (WS_SCAL + 16)                 // 1024
#define WS_RPART  (WS_READS + 1024)              // 512*256

// ---------------- device helpers ----------------
__device__ __forceinline__ float sigmoidf_(float x) { return 1.0f / (1.0f + expf(-x)); }
__device__ __forceinline__ float softplusf_(float x) { return (x > 20.0f) ? x : log1pf(expf(x)); }

__device__ __forceinline__ float wave_sum32(float v) {
    #pragma unroll
    for (int m = 16; m > 0; m >>= 1) v += __shfl_xor(v, m, 32);
    return v;
}

// =====================================================================
// 0) concat [x ; prev_reads] -> inp (3072)
// =====================================================================
__global__ void k_concat(const float* __restrict__ x, const float* __restrict__ pr,
                         float* __restrict__ inp) {
    int i = blockIdx.x * 256 + threadIdx.x; // 0..3071
    inp[i] = (i < 2048) ? x[i] : pr[i - 2048];
}

// =====================================================================
// 1) WMMA f32 GEMV: out[n] = bias[n] + Wa[n,:]·va (+ Wb[n,:]·vb)
//    One wave per 16-output tile. V_WMMA_F32_16X16X4_F32:
//    A (16x4, broadcast vector rows), B (4x16, weight tile).
//    Lane l: n=l&15, h=l>>4.  A lane regs: (v[kb+2h], v[kb+2h+1]).
//    B lane regs: (W[nb+n, kb+2h], W[nb+n, kb+2h+1]).
//    D replicated over M -> lanes 0..15, c[0] = out value.
// =====================================================================
__global__ void k_gemv_wmma(float* __restrict__ out,
                            const float* __restrict__ Wa, const float* __restrict__ va, int Ka,
                            const float* __restrict__ Wb, const float* __restrict__ vb, int Kb,
                            const float* __restrict__ bias, int Nout) {
    int wid = (blockIdx.x * blockDim.x + threadIdx.x) >> 5;
    int ntiles = (Nout + 15) >> 4;
    if (wid >= ntiles) return;            // whole-wave uniform exit: EXEC all-1 for WMMA
    int lane = threadIdx.x & 31;
    int n = lane & 15;
    int h = lane >> 4;
    int nb = wid << 4;
    int row = nb + n; if (row >= Nout) row = Nout - 1;

    v8f c = {};
    {
        const float* wr = Wa + (size_t)row * (size_t)Ka;
        for (int kb = 0; kb < Ka; kb += 4) {
            v2f a, b;
            a.x = va[kb + 2 * h];
            a.y = va[kb + 2 * h + 1];
            // weights are single-use streams: non-temporal to keep L2 for the NTM memory
            b.x = __builtin_nontemporal_load(wr + kb + 2 * h);
            b.y = __builtin_nontemporal_load(wr + kb + 2 * h + 1);
            c = __builtin_amdgcn_wmma_f32_16x16x4_f32(false, a, false, b, (short)0, c, false, false);
        }
    }
    if (Kb > 0) {
        const float* wr = Wb + (size_t)row * (size_t)Kb;
        for (int kb = 0; kb < Kb; kb += 4) {
            v2f a, b;
            a.x = vb[kb + 2 * h];
            a.y = vb[kb + 2 * h + 1];
            b.x = __builtin_nontemporal_load(wr + kb + 2 * h);
            b.y = __builtin_nontemporal_load(wr + kb + 2 * h + 1);
            c = __builtin_amdgcn_wmma_f32_16x16x4_f32(false, a, false, b, (short)0, c, false, false);
        }
    }
    if (lane < 16) {
        int o = nb + lane;
        if (o < Nout) out[o] = c[0] + bias[o];
    }
}

// =====================================================================
// 2) LSTM pointwise: ctrl = sigmoid(f)*prev_c + sigmoid(i)*tanh(g)
// =====================================================================
__global__ void k_lstm(const float* __restrict__ gates, const float* __restrict__ prev_c,
                       float* __restrict__ ctrl) {
    int i = blockIdx.x * 256 + threadIdx.x; // 0..1023
    float ig = gates[i];
    float fg = gates[1024 + i];
    float gg = gates[2048 + i];
    ctrl[i] = sigmoidf_(fg) * prev_c[i] + sigmoidf_(ig) * tanhf(gg);
}

// =====================================================================
// 3) head params. Write head: o[0:256]=k, o[256]=beta, o[257]=g,
//    o[258:261]=shift, o[261]=gamma-1, o[262:518]=e, o[518:774]=a
//    pw: k@0, e@256, a@512, sc@768 = [beta,g,s0,s1,s2,gamma,knorm]
// =====================================================================
__global__ void k_params_w(const float* __restrict__ o, float* __restrict__ pw) {
    int t = threadIdx.x; // 256
    float kv = tanhf(o[t]);
    pw[t] = kv;
    pw[256 + t] = sigmoidf_(o[262 + t]);
    pw[512 + t] = tanhf(o[518 + t]);
    __shared__ float sm[256];
    sm[t] = kv * kv; __syncthreads();
    for (int st = 128; st > 0; st >>= 1) { if (t < st) sm[t] += sm[t + st]; __syncthreads(); }
    if (t == 0) {
        float* sc = pw + 768;
        sc[0] = softplusf_(o[256]);
        sc[1] = sigmoidf_(o[257]);
        float x0 = o[258], x1 = o[259], x2 = o[260];
        float mx = fmaxf(x0, fmaxf(x1, x2));
        float e0 = expf(x0 - mx), e1 = expf(x1 - mx), e2 = expf(x2 - mx);
        float s = e0 + e1 + e2;
        sc[2] = e0 / s; sc[3] = e1 / s; sc[4] = e2 / s;
        sc[5] = 1.0f + softplusf_(o[261]);
        sc[6] = sqrtf(sm[0]);
    }
}

__global__ void k_params_r(const float* __restrict__ o, float* __restrict__ pr) {
    int t = threadIdx.x; // 256
    float kv = tanhf(o[t]);
    pr[t] = kv;
    __shared__ float sm[256];
    sm[t] = kv * kv; __syncthreads();
    for (int st = 128; st > 0; st >>= 1) { if (t < st) sm[t] += sm[t + st]; __syncthreads(); }
    if (t == 0) {
        float* sc = pr + 256;
        sc[0] = softplusf_(o[256]);
        sc[1] = sigmoidf_(o[257]);
        float x0 = o[258], x1 = o[259], x2 = o[260];
        float mx = fmaxf(x0, fmaxf(x1, x2));
        float e0 = expf(x0 - mx), e1 = expf(x1 - mx), e2 = expf(x2 - mx);
        float s = e0 + e1 + e2;
        sc[2] = e0 / s; sc[3] = e1 / s; sc[4] = e2 / s;
        sc[5] = 1.0f + softplusf_(o[261]);
        sc[6] = sqrtf(sm[0]);
    }
}

// =====================================================================
// 4) similarity pass: sim[r] = (mem_r·k)/(||mem_r||*||k||+eps), block max
//    wave-per-row, lane owns 8 cols
// =====================================================================
__global__ void k_sim(const float* __restrict__ mem, const float* __restrict__ kvec,
                      const float* __restrict__ sc, float* __restrict__ sim,
                      float* __restrict__ blkmax) {
    int lane = threadIdx.x & 31;
    int base = lane * 8;
    float4 k0 = *(const float4*)(kvec + base);
    float4 k1 = *(const float4*)(kvec + base + 4);
    float knorm = sc[6];
    int wid = (blockIdx.x * blockDim.x + threadIdx.x) >> 5;
    int nw = (gridDim.x * blockDim.x) >> 5;
    float lmax = -3.4e38f;
    for (int r = wid; r < NROWS; r += nw) {
        const float4* rs = (const float4*)(mem + (size_t)r * MCOLS);
        float4 m0 = rs[lane * 2];
        float4 m1 = rs[lane * 2 + 1];
        float dot = m0.x * k0.x + m0.y * k0.y + m0.z * k0.z + m0.w * k0.w
                  + m1.x * k1.x + m1.y * k1.y + m1.z * k1.z + m1.w * k1.w;
        float nsq = m0.x * m0.x + m0.y * m0.y + m0.z * m0.z + m0.w * m0.w
                  + m1.x * m1.x + m1.y * m1.y + m1.z * m1.z + m1.w * m1.w;
        dot = wave_sum32(dot);
        nsq = wave_sum32(nsq);
        float s = dot / (sqrtf(nsq) * knorm + EPSF);
        if (lane == 0) sim[r] = s;
        lmax = fmaxf(lmax, s);
    }
    __shared__ float sm[256];
    sm[threadIdx.x] = lmax; __syncthreads();
    for (int st = 128; st > 0; st >>= 1) {
        if (threadIdx.x < st) sm[threadIdx.x] = fmaxf(sm[threadIdx.x], sm[threadIdx.x + st]);
        __syncthreads();
    }
    if (threadIdx.x == 0) blkmax[blockIdx.x] = sm[0];
}

// =====================================================================
// 5) fused memory update + read-key similarity:
//    mem' = mem*(1-ww⊗e) + ww⊗a ; sim_r from mem' with read key
// =====================================================================
__global__ void k_upd(const float* __restrict__ src, float* __restrict__ dst,
                      const float* __restrict__ wvec, const float* __restrict__ evec,
                      const float* __restrict__ avec, const float* __restrict__ kvec,
                      const float* __restrict__ sc, float* __restrict__ sim,
                      float* __restrict__ blkmax) {
    int lane = threadIdx.x & 31;
    int base = lane * 8;
    float4 e0 = *(const float4*)(evec + base);
    float4 e1 = *(const float4*)(evec + base + 4);
    float4 a0 = *(const float4*)(avec + base);
    float4 a1 = *(const float4*)(avec + base + 4);
    float4 k0 = *(const float4*)(kvec + base);
    float4 k1 = *(const float4*)(kvec + base + 4);
    float knorm = sc[6];
    int wid = (blockIdx.x * blockDim.x + threadIdx.x) >> 5;
    int nw = (gridDim.x * blockDim.x) >> 5;
    float lmax = -3.4e38f;
    for (int r = wid; r < NROWS; r += nw) {
        float wwi = wvec[r];
        const float4* rs = (const float4*)(src + (size_t)r * MCOLS);
        float4* rd = (float4*)(dst + (size_t)r * MCOLS);
        float4 m0 = rs[lane * 2];
        float4 m1 = rs[lane * 2 + 1];
        m0.x = m0.x * (1.0f - wwi * e0.x) + wwi * a0.x;
        m0.y = m0.y * (1.0f - wwi * e0.y) + wwi * a0.y;
        m0.z = m0.z * (1.0f - wwi * e0.z) + wwi * a0.z;
        m0.w = m0.w * (1.0f - wwi * e0.w) + wwi * a0.w;
        m1.x = m1.x * (1.0f - wwi * e1.x) + wwi * a1.x;
        m1.y = m1.y * (1.0f - wwi * e1.y) + wwi * a1.y;
        m1.z = m1.z * (1.0f - wwi * e1.z) + wwi * a1.z;
        m1.w = m1.w * (1.0f - wwi * e1.w) + wwi * a1.w;
        rd[lane * 2] = m0;
        rd[lane * 2 + 1] = m1;
        float dot = m0.x * k0.x + m0.y * k0.y + m0.z * k0.z + m0.w * k0.w
                  + m1.x * k1.x + m1.y * k1.y + m1.z * k1.z + m1.w * k1.w;
        float nsq = m0.x * m0.x + m0.y * m0.y + m0.z * m0.z + m0.w * m0.w
                  + m1.x * m1.x + m1.y * m1.y + m1.z * m1.z + m1.w * m1.w;
        dot = wave_sum32(dot);
        nsq = wave_sum32(nsq);
        float s = dot / (sqrtf(nsq) * knorm + EPSF);
        if (lane == 0) sim[r] = s;
        lmax = fmaxf(lmax, s);
    }
    __shared__ float sm[256];
    sm[threadIdx.x] = lmax; __syncthreads();
    for (int st = 128; st > 0; st >>= 1) {
        if (threadIdx.x < st) sm[threadIdx.x] = fmaxf(sm[threadIdx.x], sm[threadIdx.x + st]);
        __syncthreads();
    }
    if (threadIdx.x == 0) blkmax[blockIdx.x] = sm[0];
}

// =====================================================================
// 6) reductions (deterministic, single block)
// =====================================================================
__global__ void k_redmax(const float* __restrict__ in, int n, float* __restrict__ out) {
    __shared__ float sm[256];
    float v = -3.4e38f;
    for (int i = threadIdx.x; i < n; i += 256) v = fmaxf(v, in[i]);
    sm[threadIdx.x] = v; __syncthreads();
    for (int st = 128; st > 0; st >>= 1) {
        if (threadIdx.x < st) sm[threadIdx.x] = fmaxf(sm[threadIdx.x], sm[threadIdx.x + st]);
        __syncthreads();
    }
    if (threadIdx.x == 0) out[0] = sm[0];
}

__global__ void k_redsum(const float* __restrict__ in, int n, float* __restrict__ out) {
    __shared__ float sm[256];
    float v = 0.0f;
    for (int i = threadIdx.x; i < n; i += 256) v += in[i];
    sm[threadIdx.x] = v; __syncthreads();
    for (int st = 128; st > 0; st >>= 1) {
        if (threadIdx.x < st) sm[threadIdx.x] += sm[threadIdx.x + st];
        __syncthreads();
    }
    if (threadIdx.x == 0) out[0] = sm[0];
}

// =====================================================================
// 7) p_i = exp(beta*(sim_i - max)); block partial sums. 64 blocks x 256 x 4.
// =====================================================================
__global__ void k_exp(const float* __restrict__ sim, const float* __restrict__ sc,
                      const float* __restrict__ mxp, float* __restrict__ p,
                      float* __restrict__ bsum) {
    float beta = sc[0];
    float mx = mxp[0];
    float local = 0.0f;
    int base = blockIdx.x * 1024 + threadIdx.x;
    #pragma unroll
    for (int i = 0; i < 4; ++i) {
        int idx = base + i * 256;
        float v = expf(beta * (sim[idx] - mx));
        p[idx] = v;
        local += v;
    }
    __shared__ float sm[256];
    sm[threadIdx.x] = local; __syncthreads();
    for (int st = 128; st > 0; st >>= 1) {
        if (threadIdx.x < st) sm[threadIdx.x] += sm[threadIdx.x + st];
        __syncthreads();
    }
    if (threadIdx.x == 0) bsum[blockIdx.x] = sm[0];
}

// =====================================================================
// 8) interpolate + circular shift + sharpen; wg computed on the fly:
//    wg(j) = g*p[j]/S + (1-g)*prev[j]
//    wsh_i = max(s0*wg(i+1)+s1*wg(i)+s2*wg(i-1), 1e-12)^gamma ; partial sums
// =====================================================================
__global__ void k_sharpen(const float* __restrict__ p, const float* __restrict__ prevw,
                          const float* __restrict__ sc, const float* __restrict__ Sp,
                          float* __restrict__ wsh, float* __restrict__ bsum) {
    float g = sc[1], s0 = sc[2], s1 = sc[3], s2 = sc[4], gam = sc[5];
    float S = Sp[0];
    float local = 0.0f;
    int base = blockIdx.x * 1024 + threadIdx.x;
    #pragma unroll
    for (int i = 0; i < 4; ++i) {
        int idx = base + i * 256;
        int jp = (idx == NROWS - 1) ? 0 : idx + 1;
        int jm = (idx == 0) ? NROWS - 1 : idx - 1;
        float wgp = g * (p[jp] / S) + (1.0f - g) * prevw[jp];
        float wgc = g * (p[idx] / S) + (1.0f - g) * prevw[idx];
        float wgm = g * (p[jm] / S) + (1.0f - g) * prevw[jm];
        float wsv = s0 * wgp + s1 * wgc + s2 * wgm;
        float v = powf(fmaxf(wsv, 1e-12f), gam);
        wsh[idx] = v;
        local += v;
    }
    __shared__ float sm[256];
    sm[threadIdx.x] = local; __syncthreads();
    for (int st = 128; st > 0; st >>= 1) {
        if (threadIdx.x < st) sm[threadIdx.x] += sm[threadIdx.x + st];
        __syncthreads();
    }
    if (threadIdx.x == 0) bsum[blockIdx.x] = sm[0];
}

// =====================================================================
// 9) normalize: w_i = wsh_i / (T + eps)
// =====================================================================
__global__ void k_norm(const float* __restrict__ wsh, const float* __restrict__ Tp,
                       float* __restrict__ w) {
    float inv = 1.0f / (Tp[0] + EPSF);
    int base = blockIdx.x * 1024 + threadIdx.x;
    #pragma unroll
    for (int i = 0; i < 4; ++i) {
        int idx = base + i * 256;
        w[idx] = wsh[idx] * inv;
    }
}

// =====================================================================
// 10) weighted read: reads = rw @ mem (column sums). 512 blocks x 128 rows.
// =====================================================================
__global__ void k_read_part(const float* __restrict__ mem, const float* __restrict__ wvec,
                            float* __restrict__ part) {
    int t = threadIdx.x;           // column
    int b = blockIdx.x;
    float acc = 0.0f;
    int r0 = b * 128;
    for (int r = r0; r < r0 + 128; ++r)
        acc += wvec[r] * mem[(size_t)r * MCOLS + t];
    part[(size_t)b * MCOLS + t] = acc;
}

__global__ void k_read_final(const float* __restrict__ part, float* __restrict__ reads) {
    int t = threadIdx.x;
    float acc = 0.0f;
    for (int b = 0; b < 512; ++b) acc += part[(size_t)b * MCOLS + t];
    reads[t] = acc;
}

// =====================================================================
// host side
// =====================================================================
extern "C" void kernel_launch(void* const* d_in, const int* in_sizes, int n_in,
                              void* d_out, int out_size, void* d_ws, size_t ws_size,
                              hipStream_t stream) {
    (void)in_sizes; (void)n_in; (void)out_size; (void)ws_size;
    const float* x          = (const float*)d_in[0];
    const float* prev_reads = (const float*)d_in[1];
    const float* prev_h     = (const float*)d_in[2];
    const float* prev_c     = (const float*)d_in[3];
    const float* memory     = (const float*)d_in[4];
    const float* prev_rw    = (const float*)d_in[5];
    const float* prev_ww    = (const float*)d_in[6];
    const float* W_ih       = (const float*)d_in[7];
    const float* W_hh       = (const float*)d_in[8];
    const float* b_lstm     = (const float*)d_in[9];
    const float* W_out      = (const float*)d_in[10];
    const float* b_out      = (const float*)d_in[11];
    const float* Ww         = (const float*)d_in[12];
    const float* bw         = (const float*)d_in[13];
    const float* Wr         = (const float*)d_in[14];
    const float* br         = (const float*)d_in[15];
    float* ws  = (float*)d_ws;
    float* out = (float*)d_out;

    // controller
    k_concat<<<12, 256, 0, stream>>>(x, prev_reads, ws + WS_INP);
    k_gemv_wmma<<<32, 256, 0, stream>>>(ws + WS_GATES, W_ih, ws + WS_INP, 3072,
                                        W_hh, prev_h, 1024, b_lstm, 4096);
    k_lstm<<<4, 256, 0, stream>>>(ws + WS_GATES, prev_c, ws + WS_CTRL);

    for (int hd = 0; hd < HHEADS; ++hd) {
        const float* msrc = (hd == 0) ? memory : (ws + WS_MEM);

        // head fully-connected outputs (WMMA f32 GEMV)
        k_gemv_wmma<<<7, 256, 0, stream>>>(ws + WS_OW, Ww + (size_t)hd * 774 * 1024,
                                           ws + WS_CTRL, 1024, nullptr, nullptr, 0,
                                           bw + (size_t)hd * 774, 774);
        k_params_w<<<1, 256, 0, stream>>>(ws + WS_OW, ws + WS_PW);
        k_gemv_wmma<<<3, 256, 0, stream>>>(ws + WS_ORR, Wr + (size_t)hd * 262 * 1024,
                                           ws + WS_CTRL, 1024, nullptr, nullptr, 0,
                                           br + (size_t)hd * 262, 262);
        k_params_r<<<1, 256, 0, stream>>>(ws + WS_ORR, ws + WS_PR);

        // ---- write-head addressing ----
        k_sim<<<1024, 256, 0, stream>>>(msrc, ws + WS_PW, ws + WS_PW + 768,
                                        ws + WS_SIM, ws + WS_BLK);
        k_redmax<<<1, 256, 0, stream>>>(ws + WS_BLK, 1024, ws + WS_SCAL + 0);
        k_exp<<<64, 256, 0, stream>>>(ws + WS_SIM, ws + WS_PW + 768, ws + WS_SCAL + 0,
                                      ws + WS_P, ws + WS_BLK);
        k_redsum<<<1, 256, 0, stream>>>(ws + WS_BLK, 64, ws + WS_SCAL + 1);
        k_sharpen<<<64, 256, 0, stream>>>(ws + WS_P, prev_ww + (size_t)hd * NROWS,
                                          ws + WS_PW + 768, ws + WS_SCAL + 1,
                                          ws + WS_WSH, ws + WS_BLK);
        k_redsum<<<1, 256, 0, stream>>>(ws + WS_BLK, 64, ws + WS_SCAL + 2);
        k_norm<<<64, 256, 0, stream>>>(ws + WS_WSH, ws + WS_SCAL + 2, ws + WS_WVEC);

        // ---- memory update fused with read-key similarity ----
        k_upd<<<1024, 256, 0, stream>>>(msrc, ws + WS_MEM, ws + WS_WVEC,
                                        ws + WS_PW + 256, ws + WS_PW + 512,
                                        ws + WS_PR, ws + WS_PR + 256,
                                        ws + WS_SIM, ws + WS_BLK);

        // ---- read-head addressing ----
        k_redmax<<<1, 256, 0, stream>>>(ws + WS_BLK, 1024, ws + WS_SCAL + 0);
        k_exp<<<64, 256, 0, stream>>>(ws + WS_SIM, ws + WS_PR + 256, ws + WS_SCAL + 0,
                                      ws + WS_P, ws + WS_BLK);
        k_redsum<<<1, 256, 0, stream>>>(ws + WS_BLK, 64, ws + WS_SCAL + 1);
        k_sharpen<<<64, 256, 0, stream>>>(ws + WS_P, prev_rw + (size_t)hd * NROWS,
                                          ws + WS_PR + 256, ws + WS_SCAL + 1,
                                          ws + WS_WSH, ws + WS_BLK);
        k_redsum<<<1, 256, 0, stream>>>(ws + WS_BLK, 64, ws + WS_SCAL + 2);
        k_norm<<<64, 256, 0, stream>>>(ws + WS_WSH, ws + WS_SCAL + 2, ws + WS_WVEC);

        // ---- weighted read ----
        k_read_part<<<512, 256, 0, stream>>>(ws + WS_MEM, ws + WS_WVEC, ws + WS_RPART);
        k_read_final<<<1, 256, 0, stream>>>(ws + WS_RPART, ws + WS_READS + (size_t)hd * 256);
    }

    // output projection
    k_gemv_wmma<<<16, 256, 0, stream>>>(out, W_out, ws + WS_READS, 1024,
                                        nullptr, nullptr, 0, b_out, 2048);
}